// WinogradConvModule_26774826123743
// MI455X (gfx1250) — compile-verified
//
#include <hip/hip_runtime.h>
#include <hip/hip_bf16.h>

typedef __attribute__((ext_vector_type(2))) float v2f;
typedef __attribute__((ext_vector_type(4))) float v4f;

#define C_DIM      1024
#define N_IN       8192
#define N_OUT      8190
#define NTHREADS   256
#define PER_THREAD 32   // outputs per thread (256*32 = 8192 >= 8190)

// Depthwise K=3 valid conv, one (b,c) row per block.
// Stage the whole row into LDS with CDNA5 async global->LDS b128 loads,
// then compute 3-tap FMAs out of LDS and stream NT stores to HBM.
__global__ __launch_bounds__(NTHREADS) void dwconv3_async_kernel(
    const float* __restrict__ x,
    const float* __restrict__ filt,
    float* __restrict__ out) {
  __shared__ float s[N_IN + 4];  // +4 pad so the last v4f read stays in-bounds

  const int row = blockIdx.x;           // row = b*C + c
  const int c   = row & (C_DIM - 1);
  const int t   = threadIdx.x;

  // ---------------- async global -> LDS stage ----------------
  // 8 x b128 per thread, fully coalesced: 256 threads * 16B = 4 KB per pass.
  const float* src = x + (size_t)row * N_IN;
  const unsigned lds_base = (unsigned)(uintptr_t)&s[0];  // low 32b = LDS offset

#pragma unroll
  for (int i = 0; i < 8; ++i) {
    unsigned voff = (unsigned)(t * 16 + i * (NTHREADS * 16));
    unsigned ldso = lds_base + voff;
    asm volatile("global_load_async_to_lds_b128 %0, %1, %2 th:TH_LOAD_NT"
                 :
                 : "v"(ldso), "v"(voff), "s"(src)
                 : "memory");
  }

  // Per-channel taps (uniform across block -> scalar loads).
  const float f0 = filt[c * 3 + 0];
  const float f1 = filt[c * 3 + 1];
  const float f2 = filt[c * 3 + 2];

  asm volatile("s_wait_asynccnt 0x0" ::: "memory");
  __syncthreads();

  // ---------------- compute phase ----------------
  const int obase = t * PER_THREAD;
  float* drow = out + (size_t)row * N_OUT;

  v4f vcur = *(const v4f*)&s[obase];  // 128-bit aligned LDS load

  // Iterations 0..6: all stores provably in-range (max index 8187 < 8190).
#pragma unroll
  for (int j = 0; j < PER_THREAD / 4 - 1; ++j) {
    const int o = obase + 4 * j;
    v4f vnext = *(const v4f*)&s[o + 4];

    v2f r0, r1;
    r0.x = fmaf(vcur.x, f0, fmaf(vcur.y, f1, vcur.z * f2));
    r0.y = fmaf(vcur.y, f0, fmaf(vcur.z, f1, vcur.w * f2));
    r1.x = fmaf(vcur.z, f0, fmaf(vcur.w, f1, vnext.x * f2));
    r1.y = fmaf(vcur.w, f0, fmaf(vnext.x, f1, vnext.y * f2));

    // Row base is only 8B aligned (8190*4 % 16 != 0) -> 64-bit NT stores.
    __builtin_nontemporal_store(r0, (v2f*)(drow + o));
    __builtin_nontemporal_store(r1, (v2f*)(drow + o + 2));

    vcur = vnext;
  }

  // Final iteration (j = 7): only the second pair of thread 255 can overflow
  // (outputs 8190,8191) -> single guard for the whole kernel.
  {
    const int o = obase + PER_THREAD - 4;       // obase + 28
    v4f vnext = *(const v4f*)&s[o + 4];

    v2f r0, r1;
    r0.x = fmaf(vcur.x, f0, fmaf(vcur.y, f1, vcur.z * f2));
    r0.y = fmaf(vcur.y, f0, fmaf(vcur.z, f1, vcur.w * f2));
    r1.x = fmaf(vcur.z, f0, fmaf(vcur.w, f1, vnext.x * f2));
    r1.y = fmaf(vcur.w, f0, fmaf(vnext.x, f1, vnext.y * f2));

    __builtin_nontemporal_store(r0, (v2f*)(drow + o));   // <= 8188..8189: valid
    if (o + 2 < N_OUT)
      __builtin_nontemporal_store(r1, (v2f*)(drow + o + 2));
  }
}

extern "C" void kernel_launch(void* const* d_in, const int* in_sizes, int n_in,
                              void* d_out, int out_size, void* d_ws, size_t ws_size,
                              hipStream_t stream) {
  const float* x    = (const float*)d_in[0];   // (8, 1024, 8192) fp32
  const float* filt = (const float*)d_in[1];   // (1024, 3) fp32
  float* out        = (float*)d_out;           // (8, 1024, 8190) fp32

  const int rows = in_sizes[0] / N_IN;         // B*C = 8192
  dwconv3_async_kernel<<<rows, NTHREADS, 0, stream>>>(x, filt, out);
}